// Resample_20512763805843
// MI455X (gfx1250) — compile-verified
//
#include <hip/hip_runtime.h>
#include <hip/hip_bf16.h>

// Resample (linear interpolation gather):
//   out[b,t,:] = (1-w)*original[b,k0,:] + w*original[b,k1,:]
//   gs = gamma[b,t]*(L-1); k0 = clip(floor(gs),0,L-1); k1 = min(k0+1,L-1); w = gs-k0
//
// Shapes fixed by the reference harness:
//   gamma: [B,T] = [8,8192] f32, original: [B,L,D] = [8,4096,256] f32,
//   out:   [B,T,D] f32.
//
// Design (MI455X, gfx1250, wave32):
//   * 1 thread = 1 float4 of one output row -> global_load_b128 x2 + store_b128.
//   * 64 lanes per (b,t) row; blockDim=256 (multiple of 64) => every wave32 is
//     inside a single row => gamma is wave-uniform; readfirstlane lets the
//     compiler emit a scalar load for it.
//   * Output is streamed once: non-temporal store (th:TH_NT) so the 67MB
//     stream doesn't evict the L2-resident 33.5MB gather source.
//   * No WMMA: 3 flops per output element, ~0.1 flop/byte -> pure bandwidth
//     kernel, floor ~4.3us at 23.3 TB/s.

typedef float v4f __attribute__((ext_vector_type(4)));

namespace {
constexpr int kB = 8;
constexpr int kT = 8192;
constexpr int kL = 4096;
constexpr int kD = 256;
constexpr int kVec = 4;                 // float4 per thread
constexpr int kVecsPerRow = kD / kVec;  // 64 lanes per (b,t) row
constexpr long long kTotalVecs = (long long)kB * kT * kVecsPerRow;  // 4,194,304
}

__global__ __launch_bounds__(256) void resample_lerp_kernel(
    const float* __restrict__ gamma,
    const float* __restrict__ original,
    float* __restrict__ out) {
  const long long i = (long long)blockIdx.x * blockDim.x + threadIdx.x;
  if (i >= kTotalVecs) return;

  // i -> (row = b*T + t, v = float4 slot within the D=256 row)
  const int row = (int)(i >> 6);        // i / kVecsPerRow
  const int v   = (int)(i & (kVecsPerRow - 1));
  const int b   = row >> 13;            // row / kT

  // row is uniform across the wave (64 lanes per row, 32 lanes per wave,
  // blockDim % 64 == 0). readfirstlane makes uniformity explicit so the
  // gamma fetch can become an s_load (scalar cache) instead of 32 VMEM lanes.
  const int srow = __builtin_amdgcn_readfirstlane(row);
  const float g  = gamma[srow];

  // Interpolation coordinates, faithful to the reference clipping.
  const float gs = g * (float)(kL - 1);
  int k0 = (int)floorf(gs);
  k0 = k0 < 0 ? 0 : (k0 > kL - 1 ? kL - 1 : k0);
  const int k1 = (k0 + 1 < kL - 1) ? (k0 + 1) : (kL - 1);
  const float w  = gs - (float)k0;
  const float w0 = 1.0f - w;

  // Gather two contiguous 1KB rows; 64 consecutive lanes cover a row, so each
  // wave32 issues fully-coalesced b128 loads over a contiguous 512B span.
  const long long base = ((long long)b * kL) * kD;
  const v4f* __restrict__ r0 =
      (const v4f*)(original + base + (long long)k0 * kD);
  const v4f* __restrict__ r1 =
      (const v4f*)(original + base + (long long)k1 * kD);
  const v4f a = r0[v];
  const v4f c = r1[v];

  v4f o;
  o.x = w0 * a.x + w * c.x;
  o.y = w0 * a.y + w * c.y;
  o.z = w0 * a.z + w * c.z;
  o.w = w0 * a.w + w * c.w;

  // Write-once 67MB stream: non-temporal store keeps L2 for the gather source.
  __builtin_nontemporal_store(o, (v4f*)out + i);
}

extern "C" void kernel_launch(void* const* d_in, const int* in_sizes, int n_in,
                              void* d_out, int out_size, void* d_ws, size_t ws_size,
                              hipStream_t stream) {
  (void)in_sizes; (void)n_in; (void)out_size; (void)d_ws; (void)ws_size;
  const float* gamma    = (const float*)d_in[0];
  const float* original = (const float*)d_in[1];
  float* out            = (float*)d_out;

  const int block = 256;  // 8 waves of wave32; 64 lanes per (b,t) row
  const long long grid = (kTotalVecs + block - 1) / block;  // 16384 blocks
  resample_lerp_kernel<<<(dim3)(unsigned)grid, block, 0, stream>>>(
      gamma, original, out);
}